// ContinuousBlock_9165460210338
// MI455X (gfx1250) — compile-verified
//
#include <hip/hip_runtime.h>

#define GROUPS 8
#define DG     32
#define RNK    16
#define KCH    8      // Chebyshev degree -> 9 coefficients
#define BATCH  256
#define WLEN   512
#define DIM    256
#define DT     0.1f
#define CSTR   36     // padded LDS row stride in floats (144B, 16B aligned, bank-conflict free)

typedef __attribute__((ext_vector_type(2))) float v2f;
typedef __attribute__((ext_vector_type(8))) float v8f;

// ---------------------------------------------------------------------------
// Phase 0: build per-group propagator  M_g = (I + dt*A + 0.5*dt^2*A^2)^4
// where A = U V^T - V U^T - S S^T + (tr(S S^T)/dg) I   (32x32 per group)
// One block per group, 1024 threads = one thread per matrix entry.
// ---------------------------------------------------------------------------
__global__ __launch_bounds__(1024) void build_propagator(
    const float* __restrict__ U, const float* __restrict__ V,
    const float* __restrict__ S, float* __restrict__ Mout)
{
    __shared__ float A[DG * DG];
    __shared__ float Pm[DG * DG];
    __shared__ float trR_s;

    const int g = blockIdx.x;
    const int t = threadIdx.x;          // 0..1023
    const int d = t >> 5;
    const int e = t & 31;

    const float* Ug = U + g * DG * RNK;
    const float* Vg = V + g * DG * RNK;
    const float* Sg = S + g * DG * RNK;

    float a = 0.f;
    #pragma unroll
    for (int r = 0; r < RNK; ++r) {
        a += Ug[d * RNK + r] * Vg[e * RNK + r]
           - Vg[d * RNK + r] * Ug[e * RNK + r]
           - Sg[d * RNK + r] * Sg[e * RNK + r];
    }
    A[t] = a;
    if (t == 0) {
        float s = 0.f;
        for (int i = 0; i < DG * RNK; ++i) s += Sg[i] * Sg[i];
        trR_s = s;
    }
    __syncthreads();
    if (d == e) A[t] = a + trR_s * (1.f / (float)DG);
    __syncthreads();

    // P = I + dt*A + 0.5*dt^2 * A^2
    float aa = 0.f;
    #pragma unroll
    for (int k = 0; k < DG; ++k) aa += A[d * DG + k] * A[k * DG + e];
    Pm[t] = (d == e ? 1.f : 0.f) + DT * A[t] + (0.5f * DT * DT) * aa;
    __syncthreads();

    // Q = P^2  (into A; A is dead after this point's reads, all past barrier)
    float q = 0.f;
    #pragma unroll
    for (int k = 0; k < DG; ++k) q += Pm[d * DG + k] * Pm[k * DG + e];
    A[t] = q;
    __syncthreads();

    // M = Q^2 = P^4
    float m = 0.f;
    #pragma unroll
    for (int k = 0; k < DG; ++k) m += A[d * DG + k] * A[k * DG + e];
    Mout[g * DG * DG + t] = m;
}

// ---------------------------------------------------------------------------
// Phase 1: fused group-norm -> Chebyshev filter (recursion along W) ->
//          propagator GEMM  Z = Y * M_g^T  via V_WMMA_F32_16X16X4_F32.
// One block per (batch b, group g): tile [512 rows, 32 channels].
// 256 threads = 8 wave32; each wave covers exactly one row per i-step, so
// per-token group statistics are wave shuffles.
// ---------------------------------------------------------------------------
__global__ __launch_bounds__(256) void cheb_ph_kernel(
    const float* __restrict__ X, const float* __restrict__ Cheb,
    const float* __restrict__ Mprop, float* __restrict__ Out)
{
    __shared__ float bufA[WLEN * CSTR];   // 73728 B
    __shared__ float bufB[WLEN * CSTR];   // 73728 B   (total 147456 B)

    const int blk  = blockIdx.x;
    const int b    = blk >> 3;
    const int g    = blk & 7;
    const int t    = threadIdx.x;
    const int lane = t & 31;      // channel within group; wmma lane
    const int wv   = t >> 5;      // wave id 0..7

    // Chebyshev coefficients for this group (uniform -> scalar loads)
    float c[KCH + 1];
    #pragma unroll
    for (int k = 0; k <= KCH; ++k) c[k] = Cheb[g * (KCH + 1) + k];

    const size_t base = ((size_t)b * WLEN) * DIM + (size_t)g * DG;

    float yacc[64];   // y accumulator: thread owns rows w = 8*i + wv, col = lane

    // ---- Pass 1: load + groupwise norm -> bufA (T0);  y = c0*T0
    #pragma unroll
    for (int i = 0; i < 64; ++i) {
        const int w = 8 * i + wv;
        float x = __builtin_nontemporal_load(&X[base + (size_t)w * DIM + lane]);
        float s = x;
        #pragma unroll
        for (int off = 16; off >= 1; off >>= 1) s += __shfl_xor(s, off, 32);
        const float mu = s * (1.f / 32.f);
        const float dv = x - mu;
        float vv = dv * dv;
        #pragma unroll
        for (int off = 16; off >= 1; off >>= 1) vv += __shfl_xor(vv, off, 32);
        const float xn = dv * rsqrtf(vv * (1.f / 32.f) + 1e-6f);
        bufA[w * CSTR + lane] = xn;
        yacc[i] = c[0] * xn;
    }
    __syncthreads();

    // ---- Pass 2: T1 = Lhat(T0) = -0.5*(left+right) -> bufB;  y += c1*T1
    #pragma unroll
    for (int i = 0; i < 64; ++i) {
        const int w = 8 * i + wv;
        const float l = (w > 0)        ? bufA[(w - 1) * CSTR + lane] : 0.f;
        const float r = (w < WLEN - 1) ? bufA[(w + 1) * CSTR + lane] : 0.f;
        const float t1 = -0.5f * (l + r);
        bufB[w * CSTR + lane] = t1;
        yacc[i] += c[1] * t1;
    }

    // ---- Passes k=2..8: T2 = 2*Lhat(T1) - T0 = -(l+r) - T0, overwrite T0 slot
    float* tprev = bufA;
    float* tcur  = bufB;
    for (int k = 2; k <= KCH; ++k) {
        __syncthreads();
        const float ck = c[k];
        #pragma unroll
        for (int i = 0; i < 64; ++i) {
            const int w = 8 * i + wv;
            const float l = (w > 0)        ? tcur[(w - 1) * CSTR + lane] : 0.f;
            const float r = (w < WLEN - 1) ? tcur[(w + 1) * CSTR + lane] : 0.f;
            const float t2 = -(l + r) - tprev[w * CSTR + lane];
            tprev[w * CSTR + lane] = t2;          // same-address RMW, no cross-thread use
            yacc[i] += ck * t2;
        }
        float* tmp = tprev; tprev = tcur; tcur = tmp;
    }
    __syncthreads();

    // ---- Dump y into bufA as the WMMA A-operand matrix Y[512,32]
    #pragma unroll
    for (int i = 0; i < 64; ++i) {
        const int w = 8 * i + wv;
        bufA[w * CSTR + lane] = yacc[i];
    }

    // ---- B fragments: B(kk,n) = M_g^T[e0+kk, d0+n] = M_g[16j+n, 4q+kk],
    //      kk = 2*(lane>>4) + vgpr   (mirrors documented A 16x4 layout)
    const float* Mg  = Mprop + g * DG * DG;
    const int half = lane >> 4;
    const int n15  = lane & 15;
    v2f bf[2][8];
    #pragma unroll
    for (int j = 0; j < 2; ++j) {
        #pragma unroll
        for (int q = 0; q < 8; ++q) {
            const int row = 16 * j + n15;
            bf[j][q].x = Mg[row * DG + 4 * q + 2 * half + 0];
            bf[j][q].y = Mg[row * DG + 4 * q + 2 * half + 1];
        }
    }
    __syncthreads();

    // ---- GEMM: each wave does 4 w-blocks x 2 d-blocks of 16x16 output,
    //      K=32 accumulated as 8 chunks of V_WMMA_F32_16X16X4_F32.
    #pragma unroll
    for (int tile = 0; tile < 4; ++tile) {
        const int w0 = (wv + 8 * tile) * 16;
        #pragma unroll
        for (int j = 0; j < 2; ++j) {
            v8f acc = {};
            #pragma unroll
            for (int q = 0; q < 8; ++q) {
                v2f a;
                a.x = bufA[(w0 + n15) * CSTR + 4 * q + 2 * half + 0];
                a.y = bufA[(w0 + n15) * CSTR + 4 * q + 2 * half + 1];
                acc = __builtin_amdgcn_wmma_f32_16x16x4_f32(
                          false, a, false, bf[j][q], (short)0, acc, false, false);
            }
            // C/D layout: vgpr jj, lane -> row = jj + 8*half, col = n15
            #pragma unroll
            for (int jj = 0; jj < 8; ++jj) {
                __builtin_nontemporal_store(
                    acc[jj],
                    &Out[base + (size_t)(w0 + jj + 8 * half) * DIM + 16 * j + n15]);
            }
        }
    }
}

// ---------------------------------------------------------------------------
extern "C" void kernel_launch(void* const* d_in, const int* in_sizes, int n_in,
                              void* d_out, int out_size, void* d_ws, size_t ws_size,
                              hipStream_t stream) {
    (void)in_sizes; (void)n_in; (void)out_size; (void)ws_size;
    const float* x    = (const float*)d_in[0];
    const float* cheb = (const float*)d_in[1];
    const float* U    = (const float*)d_in[2];
    const float* V    = (const float*)d_in[3];
    const float* S    = (const float*)d_in[4];
    float* out = (float*)d_out;
    float* M   = (float*)d_ws;   // 8 * 32 * 32 floats = 32 KB

    build_propagator<<<GROUPS, 1024, 0, stream>>>(U, V, S, M);
    cheb_ph_kernel<<<BATCH * GROUPS, 256, 0, stream>>>(x, cheb, M, out);
}